// TrajectoryLoss_27049704030440
// MI455X (gfx1250) — compile-verified
//
#include <hip/hip_runtime.h>
#include <math.h>

// Problem shape (fixed by the reference): pred [64,900,14], gt [64,128,14],
// frames [10], out [64,900,128] f32.
#define NB    64
#define NPRED 900
#define NGT   128
#define TD    14
#define ROWS  36   // 900 = 25 * 36; 36 = 9 passes * 4 rows

typedef float v2f __attribute__((ext_vector_type(2)));

// Low 32 bits of a flat pointer to a __shared__ object == wave-relative LDS
// byte offset (aperture layout: offset is the address's low 32 bits).
__device__ __forceinline__ unsigned lds_lo32(const void* p) {
  return (unsigned)(unsigned long long)(uintptr_t)p;
}

// ---------------------------------------------------------------------------
// Phase 1: swept-box AABB per trajectory.
// max over the 4 rotated corners of (+-hl*c +- hw*s) == |hl*c| + |hw*s|.
// ---------------------------------------------------------------------------
__device__ __forceinline__ float4 traj_aabb(const float* __restrict__ t,
                                            float t0, float t1) {
  float x_e = t[0];   // X
  float y_e = t[1];   // Y
  float vx  = t[3];   // VX
  float vy  = t[4];   // VY
  float ax  = t[6];   // AX
  float ay  = t[7];   // AY
  float yaw0= t[9];   // YAW
  float l   = t[10];  // LENGTH
  float w   = t[11];  // WIDTH
  float st  = t[13];  // STATIC

  float x_s = x_e + vx * t0 + 0.5f * ax * t0 * t0;
  float y_s = y_e + vy * t0 + 0.5f * ay * t0 * t0;
  float x_m = (x_s + x_e) * 0.5f;
  float y_m = (y_s + y_e) * 0.5f;
  float tm  = (t0 + t1) * 0.5f;
  float vxm = vx + ax * tm;
  float vym = vy + ay * tm;
  float dx  = x_e - x_s;
  float dy  = y_e - y_s;
  float move  = sqrtf(dx * dx + dy * dy);
  float speed = sqrtf(vxm * vxm + vym * vym);
  bool  rot   = (speed > 0.2f) && !(st > 0.5f);
  float yaw   = rot ? atan2f(vym, vxm) : yaw0;
  float s, c;
  sincosf(yaw, &s, &c);
  float hl = 0.5f * (l + move);
  float hw = 0.5f * w;
  float ex = fabsf(hl * c) + fabsf(hw * s);
  float ey = fabsf(hl * s) + fabsf(hw * c);
  return make_float4(x_m - ex, y_m - ey, x_m + ex, y_m + ey);
}

__global__ void traj_aabb_kernel(const float* __restrict__ pred,
                                 const float* __restrict__ gt,
                                 const float* __restrict__ frames, int nf,
                                 float4* __restrict__ pbox,
                                 float4* __restrict__ gbox) {
  int i = blockIdx.x * blockDim.x + threadIdx.x;
  float t0 = frames[0];
  float t1 = frames[nf - 1];
  const int np = NB * NPRED;
  const int ng = NB * NGT;
  if (i < np) {
    pbox[i] = traj_aabb(pred + (size_t)i * TD, t0, t1);
  } else if (i < np + ng) {
    int j = i - np;
    gbox[j] = traj_aabb(gt + (size_t)j * TD, t0, t1);
  }
}

// ---------------------------------------------------------------------------
// Phase 2: pairwise IoU [B, N, M]. Store-bandwidth bound (29.5 MB out).
// gt boxes (per batch) and pred boxes (per block) staged to LDS via CDNA5
// async global->LDS DMA (ASYNCcnt path); output written with non-temporal
// b64 stores (2 adjacent gt columns per thread).
// ---------------------------------------------------------------------------
__global__ __launch_bounds__(256) void iou_kernel(
    const float4* __restrict__ pbox,
    const float4* __restrict__ gbox,
    float* __restrict__ out) {
  __shared__ float4 s_g[NGT];
  __shared__ float4 s_p[ROWS];
  __shared__ float  s_ga[NGT];
  __shared__ float  s_pa[ROWS];

  const int b    = blockIdx.y;
  const int row0 = blockIdx.x * ROWS;
  const int t    = threadIdx.x;

  // Async-stage boxes into LDS (16B DMA per active lane).
  if (t < NGT) {
    const float4* src = gbox + b * NGT + t;
    asm volatile("global_load_async_to_lds_b128 %0, %1, off"
                 :: "v"(lds_lo32(&s_g[t])),
                    "v"((unsigned long long)(uintptr_t)src)
                 : "memory");
  } else if (t < NGT + ROWS) {
    int r = t - NGT;
    const float4* src = pbox + b * NPRED + row0 + r;
    asm volatile("global_load_async_to_lds_b128 %0, %1, off"
                 :: "v"(lds_lo32(&s_p[r])),
                    "v"((unsigned long long)(uintptr_t)src)
                 : "memory");
  }
  asm volatile("s_wait_asynccnt 0x0" ::: "memory");
  __syncthreads();

  // Precompute areas once.
  if (t < NGT) {
    float4 g = s_g[t];
    s_ga[t] = (g.z - g.x) * (g.w - g.y);
  } else if (t < NGT + ROWS) {
    int r = t - NGT;
    float4 p = s_p[r];
    s_pa[r] = (p.z - p.x) * (p.w - p.y);
  }
  __syncthreads();

  // Thread owns 2 adjacent gt columns; 4 rows per pass across 256 threads.
  const int    col = (t & 63) * 2;
  const int    ty  = t >> 6;  // 0..3
  const float4 g0  = s_g[col];
  const float4 g1  = s_g[col + 1];
  const float  ga0 = s_ga[col];
  const float  ga1 = s_ga[col + 1];
  float* obase = out + ((size_t)b * NPRED + row0) * NGT + col;

#pragma unroll
  for (int r0 = 0; r0 < ROWS; r0 += 4) {
    int r = r0 + ty;
    float4 p = s_p[r];
    float pa = s_pa[r];

    float ix1 = fmaxf(p.x, g0.x);
    float iy1 = fmaxf(p.y, g0.y);
    float ix2 = fminf(p.z, g0.z);
    float iy2 = fminf(p.w, g0.w);
    float iw  = fmaxf(ix2 - ix1, 0.0f);
    float ih  = fmaxf(iy2 - iy1, 0.0f);
    float inter0 = iw * ih;
    float uni0   = fmaxf(pa + ga0 - inter0, 1e-8f);

    ix1 = fmaxf(p.x, g1.x);
    iy1 = fmaxf(p.y, g1.y);
    ix2 = fminf(p.z, g1.z);
    iy2 = fminf(p.w, g1.w);
    iw  = fmaxf(ix2 - ix1, 0.0f);
    ih  = fmaxf(iy2 - iy1, 0.0f);
    float inter1 = iw * ih;
    float uni1   = fmaxf(pa + ga1 - inter1, 1e-8f);

    v2f v;
    v.x = inter0 / uni0;
    v.y = inter1 / uni1;
    __builtin_nontemporal_store(v, (v2f*)(obase + (size_t)r * NGT));
  }
}

// ---------------------------------------------------------------------------
extern "C" void kernel_launch(void* const* d_in, const int* in_sizes, int n_in,
                              void* d_out, int out_size, void* d_ws, size_t ws_size,
                              hipStream_t stream) {
  (void)n_in; (void)out_size; (void)ws_size;
  const float* pred   = (const float*)d_in[0];
  const float* gt     = (const float*)d_in[1];
  const float* frames = (const float*)d_in[2];
  const int nf = in_sizes[2];

  float4* pbox = (float4*)d_ws;          // 64*900*16 B = 921600 B
  float4* gbox = pbox + NB * NPRED;      // 64*128*16 B = 131072 B

  const int total = NB * (NPRED + NGT);
  traj_aabb_kernel<<<dim3((total + 255) / 256), dim3(256), 0, stream>>>(
      pred, gt, frames, nf, pbox, gbox);

  iou_kernel<<<dim3(NPRED / ROWS, NB), dim3(256), 0, stream>>>(
      pbox, gbox, (float*)d_out);
}